// SREL_intra_phase1_tester_66675072303565
// MI455X (gfx1250) — compile-verified
//
#include <hip/hip_runtime.h>
#include <math.h>

#define B_    64
#define L_    256
#define M_    4
#define NSTEP 10
#define HID_  1024
#define DIN_  1280
#define OUT_RHO_OFF (B_*(NSTEP+1)*L_*2)   // 360448 floats of s_stack first

typedef float v2f __attribute__((ext_vector_type(2)));
typedef float v4f __attribute__((ext_vector_type(4)));
typedef float v8f __attribute__((ext_vector_type(8)));

// ---------------- static device scratch (capture-safe, rewritten every call) ----
__device__ float g_phi  [B_*L_];          // evolving phase
__device__ float g_sfeat[B_*2*L_];        // [B][512] : cos | sin
__device__ float g_feats[B_*M_*768];      // static per-(b,m) features (w_re|w_im|y)
__device__ float g_W1T  [DIN_*HID_];      // W1 transposed: [d][h]
__device__ float g_base [B_*M_*HID_];     // W1[:,512:]*feats + b1   (step-invariant)
__device__ float g_t    [B_*HID_];        // W1[:,:512]*s_feat       (per step)
__device__ float g_GsRe [B_*L_*M_];
__device__ float g_GsIm [B_*L_*M_];
__device__ float g_HsRe [B_*L_*M_];
__device__ float g_HsIm [B_*L_*M_];
__device__ float g_qG   [B_*M_];
__device__ float g_qH   [B_*M_];
__device__ float g_rho  [B_*M_];

// ---------------- setup kernels -------------------------------------------------
__global__ void copy_phi_k(const float* __restrict__ phi) {
    int i = blockIdx.x * blockDim.x + threadIdx.x;
    if (i < B_*L_) g_phi[i] = phi[i];
}

__global__ void transpose_w1_k(const float* __restrict__ W1) {
    int i = blockIdx.x * blockDim.x + threadIdx.x;   // i = d*HID + h
    if (i >= DIN_*HID_) return;
    int d = i / HID_, h = i - d*HID_;
    g_W1T[i] = W1[(size_t)h*DIN_ + d];
}

__global__ void build_feats_k(const float* __restrict__ w_re,
                              const float* __restrict__ w_im,
                              const float* __restrict__ y_M) {
    int i = blockIdx.x * blockDim.x + threadIdx.x;   // i = r*768 + j, r=b*4+m
    if (i >= B_*M_*768) return;
    int r = i / 768, j = i - r*768;
    int b = r >> 2, m = r & 3;
    float v;
    if (j < 256)       v = w_re[((size_t)b*L_ + j)      *M_ + m];
    else if (j < 512)  v = w_im[((size_t)b*L_ + (j-256))*M_ + m];
    else               v = y_M [(size_t)(j-512)*M_ + m];
    g_feats[i] = v;
}

// ---------------- fp32 WMMA GEMM: C[rows,1024] = A[rows,K] * BT[K,1024] (+bias) --
// one wave computes one 16x16 output tile via V_WMMA_F32_16X16X4_F32, K-loop of 4.
// MODE=0: base GEMM  (A=g_feats, K=768, BT=W1T[512:], C=g_base, +bias)
// MODE=1: step GEMM  (A=g_sfeat, K=512, BT=W1T,       C=g_t)
template <int MODE, bool HAS_BIAS>
__global__ void gemm16_k(const float* __restrict__ bias) {
    constexpr int LDA = (MODE == 0) ? 768 : 512;
    constexpr int K   = (MODE == 0) ? 768 : 512;
    const float* A  = (MODE == 0) ? g_feats : g_sfeat;
    const float* BT = (MODE == 0) ? (g_W1T + 512*HID_) : g_W1T;
    float*       C  = (MODE == 0) ? g_base : g_t;

    int wave = (blockIdx.x * blockDim.x + threadIdx.x) >> 5;
    int lane = threadIdx.x & 31;
    int tileB = wave >> 6;          // /64 col-tiles
    int tileH = wave & 63;
    int half  = lane >> 4;
    int ln    = lane & 15;
    int row   = tileB*16 + ln;      // A row for this lane
    int col   = tileH*16 + ln;      // B/C column for this lane

    const float* arow = A + (size_t)row * LDA;
    v8f acc = {};
#pragma unroll 8
    for (int k0 = 0; k0 < K; k0 += 4) {
        int ka = k0 + 2*half;       // VGPR0 holds K=0/2, VGPR1 holds K=1/3
        v2f a, b;
        a.x = arow[ka];
        a.y = arow[ka + 1];
        b.x = BT[(size_t)ka      * HID_ + col];
        b.y = BT[(size_t)(ka+1)  * HID_ + col];
        acc = __builtin_amdgcn_wmma_f32_16x16x4_f32(
                  false, a, false, b, (short)0, acc, false, false);
    }
    float bv = HAS_BIAS ? bias[col] : 0.f;   // uniform across output rows
#pragma unroll
    for (int r = 0; r < 8; ++r) {
        int orow = tileB*16 + r + 8*half;   // VGPR r: M=r (lanes<16), M=8+r (lanes>=16)
        C[(size_t)orow * HID_ + col] = acc[r] + bv;
    }
}

// ---------------- per-step kernels ----------------------------------------------
__global__ void step_s_k(float* __restrict__ out, int n) {
    int i = blockIdx.x * blockDim.x + threadIdx.x;   // i = b*L + l
    if (i >= B_*L_) return;
    int b = i >> 8, l = i & 255;
    float p = g_phi[i], s, c;
    sincosf(p, &s, &c);
    g_sfeat[b*512 + l]       = c;
    g_sfeat[b*512 + 256 + l] = s;
    size_t o = ((size_t)(b*(NSTEP+1) + n)*L_ + l) * 2;
    out[o] = c; out[o+1] = s;                         // complex64 interleave
}

__global__ void rho_k(const float* __restrict__ W2, const float* __restrict__ b2,
                      float* __restrict__ out, int n) {
    int wave = (blockIdx.x * blockDim.x + threadIdx.x) >> 5;   // wave = b*4+m
    int lane = threadIdx.x & 31;
    int b = wave >> 2, m = wave & 3;
    const float* tb = g_t    + (size_t)b    * HID_;
    const float* ba = g_base + (size_t)wave * HID_;
    float acc = 0.f;
    for (int h = lane; h < HID_; h += 32) {
        float v = tb[h] + ba[h];
        acc += W2[h] * (v > 0.f ? v : 0.f);
    }
#pragma unroll
    for (int off = 16; off; off >>= 1) acc += __shfl_xor(acc, off, 32);
    if (lane == 0) {
        float r = 1.f / (1.f + __expf(-(acc + b2[0])));
        g_rho[wave] = r;
        out[OUT_RHO_OFF + (b*NSTEP + n)*M_ + m] = r;
    }
}

// dominant kernel: stream G/H (268 MB) once, one wave per (b,l), NT b128 loads
__global__ void gshs_k(const float* __restrict__ G_re, const float* __restrict__ G_im,
                       const float* __restrict__ H_re, const float* __restrict__ H_im) {
    int wid  = (blockIdx.x * blockDim.x + threadIdx.x) >> 5;   // wid = b*256 + l
    int lane = threadIdx.x & 31;
    int b = wid >> 8;
    size_t rowoff = (size_t)wid * (L_*M_);
    const v4f* gre = (const v4f*)(G_re + rowoff);
    const v4f* gim = (const v4f*)(G_im + rowoff);
    const v4f* hre = (const v4f*)(H_re + rowoff);
    const v4f* him = (const v4f*)(H_im + rowoff);
    const float* sre = g_sfeat + b*512;
    const float* sim = sre + 256;

    v4f aGr = {}, aGi = {}, aHr = {}, aHi = {};
    for (int k = lane; k < L_; k += 32) {
        v4f gr = __builtin_nontemporal_load(gre + k);
        v4f gi = __builtin_nontemporal_load(gim + k);
        v4f hr = __builtin_nontemporal_load(hre + k);
        v4f hi = __builtin_nontemporal_load(him + k);
        float sr = sre[k], si = sim[k];
        aGr += gr*sr - gi*si;
        aGi += gr*si + gi*sr;
        aHr += hr*sr - hi*si;
        aHi += hr*si + hi*sr;
    }
    float v[16] = { aGr.x,aGr.y,aGr.z,aGr.w, aGi.x,aGi.y,aGi.z,aGi.w,
                    aHr.x,aHr.y,aHr.z,aHr.w, aHi.x,aHi.y,aHi.z,aHi.w };
#pragma unroll
    for (int i = 0; i < 16; ++i)
#pragma unroll
        for (int off = 16; off; off >>= 1) v[i] += __shfl_xor(v[i], off, 32);
    if (lane == 0) {
        ((v4f*)g_GsRe)[wid] = (v4f){v[0], v[1], v[2], v[3]};
        ((v4f*)g_GsIm)[wid] = (v4f){v[4], v[5], v[6], v[7]};
        ((v4f*)g_HsRe)[wid] = (v4f){v[8], v[9], v[10],v[11]};
        ((v4f*)g_HsIm)[wid] = (v4f){v[12],v[13],v[14],v[15]};
    }
}

__global__ void quad_k() {                       // qG,qH: Re(s^H G s) per (b,m)
    int wave = (blockIdx.x * blockDim.x + threadIdx.x) >> 5;   // b*4+m
    int lane = threadIdx.x & 31;
    int b = wave >> 2, m = wave & 3;
    float qg = 0.f, qh = 0.f;
    for (int l = lane; l < L_; l += 32) {
        float sr = g_sfeat[b*512 + l], si = g_sfeat[b*512 + 256 + l];
        int i = (b*L_ + l)*M_ + m;
        qg += sr*g_GsRe[i] + si*g_GsIm[i];
        qh += sr*g_HsRe[i] + si*g_HsIm[i];
    }
#pragma unroll
    for (int off = 16; off; off >>= 1) { qg += __shfl_xor(qg, off, 32);
                                         qh += __shfl_xor(qh, off, 32); }
    if (lane == 0) { g_qG[wave] = qg; g_qH[wave] = qh; }
}

__global__ void update_k() {                     // eta, eta_net, phi update
    int i = blockIdx.x * blockDim.x + threadIdx.x;   // b*L + l
    if (i >= B_*L_) return;
    int b = i >> 8;
    float sr = g_sfeat[b*512 + (i & 255)], si = g_sfeat[b*512 + 256 + (i & 255)];
    v4f Gr = ((const v4f*)g_GsRe)[i], Gi = ((const v4f*)g_GsIm)[i];
    v4f Hr = ((const v4f*)g_HsRe)[i], Hi = ((const v4f*)g_HsIm)[i];
    float en = 0.f;
#pragma unroll
    for (int m = 0; m < M_; ++m) {
        float imG = sr*Gi[m] - si*Gr[m];
        float imH = sr*Hi[m] - si*Hr[m];
        float eta = 2.f*imG / g_qG[b*M_ + m] - 2.f*imH / g_qH[b*M_ + m];
        en += g_rho[b*M_ + m] * eta;
    }
    g_phi[i] -= g_rho[b*M_ + 3] * en;            // reference uses last antenna's rho
}

// ---------------- driver --------------------------------------------------------
extern "C" void kernel_launch(void* const* d_in, const int* in_sizes, int n_in,
                              void* d_out, int out_size, void* d_ws, size_t ws_size,
                              hipStream_t stream) {
    (void)in_sizes; (void)n_in; (void)out_size; (void)d_ws; (void)ws_size;
    const float* phi  = (const float*)d_in[0];
    const float* w_re = (const float*)d_in[1];
    const float* w_im = (const float*)d_in[2];
    const float* y_M  = (const float*)d_in[3];
    const float* G_re = (const float*)d_in[4];
    const float* G_im = (const float*)d_in[5];
    const float* H_re = (const float*)d_in[6];
    const float* H_im = (const float*)d_in[7];
    const float* W1   = (const float*)d_in[8];
    const float* b1   = (const float*)d_in[9];
    const float* W2   = (const float*)d_in[10];
    const float* b2   = (const float*)d_in[11];
    float* out = (float*)d_out;

    copy_phi_k    <<<(B_*L_)/256,      256, 0, stream>>>(phi);
    transpose_w1_k<<<(DIN_*HID_)/256,  256, 0, stream>>>(W1);
    build_feats_k <<<(B_*M_*768)/256,  256, 0, stream>>>(w_re, w_im, y_M);
    // base = feats(256x768) x W1T[512:,:] + b1  -> 16*64 tiles / 8 waves per WG
    gemm16_k<0, true> <<<(16*64)/8,    256, 0, stream>>>(b1);

    for (int n = 0; n < NSTEP; ++n) {
        step_s_k <<<(B_*L_)/256,   256, 0, stream>>>(out, n);
        // t = sfeat(64x512) x W1T[:512,:]  -> 4*64 tiles / 8 waves per WG
        gemm16_k<1, false> <<<(4*64)/8, 256, 0, stream>>>(nullptr);
        rho_k    <<<(B_*M_)/8,     256, 0, stream>>>(W2, b2, out, n);
        gshs_k   <<<(B_*L_)/8,     256, 0, stream>>>(G_re, G_im, H_re, H_im);
        quad_k   <<<(B_*M_)/8,     256, 0, stream>>>();
        update_k <<<(B_*L_)/256,   256, 0, stream>>>();
    }
    step_s_k     <<<(B_*L_)/256,   256, 0, stream>>>(out, NSTEP);  // s_last
}